// PluginModel_36163624632550
// MI455X (gfx1250) — compile-verified
//
#include <hip/hip_runtime.h>
#include <hip/hip_bf16.h>

typedef __attribute__((ext_vector_type(2))) float v2f;
typedef __attribute__((ext_vector_type(8))) float v8f;

#define HIDDEN 256
#define QKV_DIM 512
#define HEAD_DIM 64
#define MAX_SEQ 8192
#define NUM_Q_HEADS 4
#define NUM_KV_HEADS 2
// per (b, kvh) cache plane: MAX_SEQ * HEAD_DIM
#define PLANE (MAX_SEQ * HEAD_DIM)

// ---------------------------------------------------------------------------
// Kernel 1: streaming copy of kv_cache -> updated cache (B128 loads/stores).
// 67,108,864 floats = 16,777,216 float4.
// ---------------------------------------------------------------------------
__global__ void copy_cache_kernel(const float4* __restrict__ src,
                                  float4* __restrict__ dst, long n4) {
    long i = (long)blockIdx.x * blockDim.x + threadIdx.x;
    long stride = (long)gridDim.x * blockDim.x;
    for (; i < n4; i += stride) dst[i] = src[i];
}

// ---------------------------------------------------------------------------
// Kernel 2: M=32 GEMM  D[b,n] = sum_k A[b,k] * W[n,k] (+ bias[n])
// via V_WMMA_F32_16X16X4_F32.  One wave per 16x16 output tile.
// Tile id: global wave w -> m_tile = w & 1 (M=0..15 / 16..31), n_tile = w >> 1.
// A frag (16x4 f32, 2 VGPRs): lanes 0-15 rows M, K = k+0/k+1; lanes 16-31 K = k+2/k+3.
// B frag (4x16 f32, 2 VGPRs): lanes 0-15 cols N, K = k+0/k+1; lanes 16-31 K = k+2/k+3.
// ---------------------------------------------------------------------------
__global__ void gemm32_wmma_kernel(const float* __restrict__ A,
                                   const float* __restrict__ W,
                                   const float* __restrict__ bias,
                                   float* __restrict__ D,
                                   int N, int K) {
    int gwave = (int)((blockIdx.x * blockDim.x + threadIdx.x) >> 5);
    int lane  = threadIdx.x & 31;
    int m_base = (gwave & 1) * 16;
    int n_base = (gwave >> 1) * 16;
    int halfSel = lane >> 4;       // 0: lanes 0-15, 1: lanes 16-31
    int l = lane & 15;
    int rowA = m_base + l;
    int colB = n_base + l;
    int kOff = halfSel * 2;        // K sub-offset 0 or 2

    v8f acc = {};
    const float* arow = A + (size_t)rowA * K;
    const float* wrow = W + (size_t)colB * K;
    for (int k = 0; k < K; k += 4) {
        v2f a, b;
        a.x = arow[k + kOff];
        a.y = arow[k + kOff + 1];
        b.x = wrow[k + kOff];
        b.y = wrow[k + kOff + 1];
        acc = __builtin_amdgcn_wmma_f32_16x16x4_f32(
            false, a, false, b, (short)0, acc, false, false);
    }
    // D layout: VGPR r -> M = m_base + r + 8*halfSel, N = n_base + (lane&15)
    int Nc = n_base + l;
    float bv = bias ? bias[Nc] : 0.0f;
    #pragma unroll
    for (int r = 0; r < 8; ++r) {
        int M = m_base + r + halfSel * 8;
        D[(size_t)M * N + Nc] = acc[r] + bv;
    }
}

// ---------------------------------------------------------------------------
// Kernel 3: RoPE on q,k; write q_rot to ws; scatter new k,v rows into cache.
// One block (256 threads) per batch.
// rope row: [cos(0..31) | sin(0..31)]
// ---------------------------------------------------------------------------
__global__ void rope_scatter_kernel(const float* __restrict__ qkv,
                                    const float* __restrict__ rope,
                                    const int* __restrict__ ctx_len,
                                    float* __restrict__ qrot,
                                    float* __restrict__ cache_out) {
    int b = blockIdx.x;
    int t = threadIdx.x;
    int pos = ctx_len[b];
    const float* rp = rope + (size_t)pos * HEAD_DIM;

    // q: 256 elements
    {
        int h = t >> 6, i = t & 63;
        const float* qh = qkv + (size_t)b * QKV_DIM + h * HEAD_DIM;
        float val;
        if (i < 32) val = qh[i] * rp[i] - qh[i + 32] * rp[32 + i];
        else { int i2 = i - 32; val = qh[i2] * rp[32 + i2] + qh[i] * rp[i2]; }
        qrot[(size_t)b * HIDDEN + t] = val;
    }
    // k (rope) and v (plain): 128 elements each
    if (t < NUM_KV_HEADS * HEAD_DIM) {
        int h = t >> 6, i = t & 63;
        const float* kh = qkv + (size_t)b * QKV_DIM + HIDDEN + h * HEAD_DIM;
        float kv;
        if (i < 32) kv = kh[i] * rp[i] - kh[i + 32] * rp[32 + i];
        else { int i2 = i - 32; kv = kh[i2] * rp[32 + i2] + kh[i] * rp[i2]; }
        size_t koff = ((size_t)((b * 2 + 0) * NUM_KV_HEADS + h)) * PLANE
                    + (size_t)pos * HEAD_DIM + i;
        cache_out[koff] = kv;
        float vv = qkv[(size_t)b * QKV_DIM + HIDDEN + NUM_KV_HEADS * HEAD_DIM
                       + h * HEAD_DIM + i];
        size_t voff = ((size_t)((b * 2 + 1) * NUM_KV_HEADS + h)) * PLANE
                    + (size_t)pos * HEAD_DIM + i;
        cache_out[voff] = vv;
    }
}

// ---------------------------------------------------------------------------
// Kernel 4: attention for one (batch, q-head) per block. 256 threads = 8 waves.
// Scores (<=8000) live in LDS (32 KB << 320 KB/WGP). Two-pass softmax with
// wave32 butterfly reductions, then weighted-V accumulation.
// ---------------------------------------------------------------------------
__device__ __forceinline__ float wave_sum(float v) {
    #pragma unroll
    for (int off = 16; off > 0; off >>= 1) v += __shfl_xor(v, off, 32);
    return v;
}
__device__ __forceinline__ float wave_max(float v) {
    #pragma unroll
    for (int off = 16; off > 0; off >>= 1) v = fmaxf(v, __shfl_xor(v, off, 32));
    return v;
}

__global__ void attn_kernel(const float* __restrict__ cache,
                            const float* __restrict__ qrot,
                            const int* __restrict__ ctx_len,
                            float* __restrict__ ctx_out) {
    __shared__ float sc[MAX_SEQ];
    __shared__ float red[8];
    __shared__ float bcast;
    __shared__ float vacc[8][64];

    int b = blockIdx.x >> 2;
    int h = blockIdx.x & 3;
    int kvh = h >> 1;                       // G = 2
    int pos = ctx_len[b];
    int L = pos + 1;
    int tid = threadIdx.x;
    int wave = tid >> 5, lane = tid & 31;

    const float* kbase = cache + ((size_t)((b * 2 + 0) * NUM_KV_HEADS + kvh)) * PLANE;
    const float* vbase = cache + ((size_t)((b * 2 + 1) * NUM_KV_HEADS + kvh)) * PLANE;
    float q0 = qrot[(size_t)b * HIDDEN + h * HEAD_DIM + lane];
    float q1 = qrot[(size_t)b * HIDDEN + h * HEAD_DIM + lane + 32];
    const float scale = 0.125f;             // 64^-0.5

    // Pass 1: scores into LDS + running max
    float lmax = -3.0e38f;
    for (int j = wave; j < L; j += 8) {
        float p = q0 * kbase[(size_t)j * 64 + lane]
                + q1 * kbase[(size_t)j * 64 + lane + 32];
        float s = wave_sum(p) * scale;
        if (lane == 0) sc[j] = s;
        lmax = fmaxf(lmax, s);
    }
    lmax = wave_max(lmax);
    if (lane == 0) red[wave] = lmax;
    __syncthreads();
    if (tid == 0) {
        float m = red[0];
        #pragma unroll
        for (int i = 1; i < 8; ++i) m = fmaxf(m, red[i]);
        bcast = m;
    }
    __syncthreads();
    float gmax = bcast;

    // Pass 2: exponentiate + sum
    float lsum = 0.0f;
    for (int j = tid; j < L; j += 256) {
        float p = __expf(sc[j] - gmax);
        sc[j] = p;
        lsum += p;
    }
    lsum = wave_sum(lsum);
    if (lane == 0) red[wave] = lsum;
    __syncthreads();
    if (tid == 0) {
        float s = 0.0f;
        #pragma unroll
        for (int i = 0; i < 8; ++i) s += red[i];
        bcast = s;
    }
    __syncthreads();
    float inv = 1.0f / bcast;

    // Pass 3: weighted V
    float a0 = 0.0f, a1 = 0.0f;
    for (int j = wave; j < L; j += 8) {
        float p = sc[j];
        a0 += p * vbase[(size_t)j * 64 + lane];
        a1 += p * vbase[(size_t)j * 64 + lane + 32];
    }
    vacc[wave][lane] = a0;
    vacc[wave][lane + 32] = a1;
    __syncthreads();
    if (wave == 0) {
        float s0 = 0.0f, s1 = 0.0f;
        #pragma unroll
        for (int w = 0; w < 8; ++w) { s0 += vacc[w][lane]; s1 += vacc[w][lane + 32]; }
        ctx_out[(size_t)b * HIDDEN + h * HEAD_DIM + lane] = s0 * inv;
        ctx_out[(size_t)b * HIDDEN + h * HEAD_DIM + lane + 32] = s1 * inv;
    }
}

// ---------------------------------------------------------------------------
// inputs: 0:x 1:kv_cache 2:ctx_len 3:rope 4:qkv_w 5:qkv_b 6:out_w
// d_out: [out (32*256) | updated_kv (32*2*2*8192*64)]
// ws: qkv (32*512) | qrot (32*256) | ctx (32*256)  = 128 KB
// ---------------------------------------------------------------------------
extern "C" void kernel_launch(void* const* d_in, const int* in_sizes, int n_in,
                              void* d_out, int out_size, void* d_ws, size_t ws_size,
                              hipStream_t stream) {
    const float* x        = (const float*)d_in[0];
    const float* kv_cache = (const float*)d_in[1];
    const int*   ctx_len  = (const int*)d_in[2];
    const float* rope     = (const float*)d_in[3];
    const float* qkv_w    = (const float*)d_in[4];
    const float* qkv_b    = (const float*)d_in[5];
    const float* out_w    = (const float*)d_in[6];

    float* out       = (float*)d_out;
    float* cache_out = out + 32 * HIDDEN;          // 8192 floats in
    float* ws        = (float*)d_ws;
    float* qkv_ws    = ws;                         // 16384 floats
    float* qrot_ws   = ws + 16384;                 // 8192 floats
    float* ctx_ws    = ws + 24576;                 // 8192 floats

    // 1) bulk cache copy (bandwidth-dominant: ~536 MB @ 23.3 TB/s)
    const long n4 = (long)32 * 2 * NUM_KV_HEADS * MAX_SEQ * HEAD_DIM / 4;
    copy_cache_kernel<<<4096, 256, 0, stream>>>(
        (const float4*)kv_cache, (float4*)cache_out, n4);

    // 2) QKV projection: 64 tiles -> 64 waves -> 16 blocks x 128
    gemm32_wmma_kernel<<<16, 128, 0, stream>>>(x, qkv_w, qkv_b, qkv_ws,
                                               QKV_DIM, HIDDEN);

    // 3) RoPE + scatter new k/v rows (after copy, overwrites row `pos`)
    rope_scatter_kernel<<<32, 256, 0, stream>>>(qkv_ws, rope, ctx_len,
                                                qrot_ws, cache_out);

    // 4) attention: one block per (b, q-head)
    attn_kernel<<<32 * NUM_Q_HEADS, 256, 0, stream>>>(cache_out, qrot_ws,
                                                      ctx_len, ctx_ws);

    // 5) output projection: 32 tiles -> 8 blocks x 128
    gemm32_wmma_kernel<<<8, 128, 0, stream>>>(ctx_ws, out_w, nullptr, out,
                                              HIDDEN, HIDDEN);
}